// RelativeAttentionPositions_37048387895400
// MI455X (gfx1250) — compile-verified
//
#include <hip/hip_runtime.h>

// Problem constants (match reference)
#define S_LEN   1024
#define D_DIM   64
#define MAXREL  128
#define H_NUM   16

typedef __attribute__((ext_vector_type(2))) float v2f;
typedef __attribute__((ext_vector_type(8))) float v8f;

// One block: 8 waves, each wave computes 16 rows of scores for one (b,h).
// Grid.x = (B*H) * (S/128) = 64 * 8 = 512.
__global__ __launch_bounds__(256) void relpos_scores_kernel(
    const float* __restrict__ X,     // [B*H, S, D] = [64,1024,64]
    const float* __restrict__ E,     // [257, 64] rel_keys_emb
    const float* __restrict__ bias,  // [H] = [16]
    float* __restrict__ out)         // [B*H, S, S]
{
    __shared__ float ldsE[256 * 64]; // 64 KB: rows v=0..255, XOR-swizzled

    const int tid  = threadIdx.x;
    const int lane = tid & 31;
    const int wave = tid >> 5;

    const int wg = blockIdx.x;       // 0..511
    const int bh = wg >> 3;          // 0..63
    const int h  = bh & (H_NUM - 1);
    const int s0 = (wg & 7) * 128 + wave * 16;   // first of this wave's 16 rows

    // ---- Stage E[0..255][0..63] into LDS, swizzled: elem (v,c) -> v*64 + (c ^ ((v&15)<<2)) ----
    for (int i = tid; i < 256 * 32; i += 256) {  // 8192 float2 chunks
        int v = i >> 5;
        int c = (i & 31) * 2;
        v2f e = *(const v2f*)(E + v * D_DIM + c);
        int sw = (v & 15) << 2;
        *(v2f*)(ldsE + v * 64 + (c ^ sw)) = e;
    }
    __syncthreads();

    // ---- Per-lane WMMA operand coordinates (f32 16x16x4 layout) ----
    const int row  = lane & 15;      // A: M index / B: N index / C: N index
    const int half = lane >> 4;      // lane-half
    const int koff = half * 2;       // K offset within a K=4 step

    const float* xrow = X + (size_t)(bh * S_LEN + s0 + row) * D_DIM;

    // Preload A: 16 K-steps of float2 per lane (32 VGPRs)
    v2f a[16];
#pragma unroll
    for (int k = 0; k < 16; ++k)
        a[k] = *(const v2f*)(xrow + 4 * k + koff);

    // ---- y256[row] = dot(X[row], E[256]) via manual dot + cross-half reduce ----
    float part = 0.f;
    {
        const float* e2 = E + 256 * D_DIM + half * 32;
        const float* x2 = xrow + half * 32;
#pragma unroll
        for (int j = 0; j < 32; j += 4) {
            float4 xv = *(const float4*)(x2 + j);
            float4 ev = *(const float4*)(e2 + j);
            part += xv.x * ev.x + xv.y * ev.y + xv.z * ev.z + xv.w * ev.w;
        }
    }
    const float y256v = part + __shfl_xor(part, 16, 32); // lane m holds y256[s0+(m&15)]

    const float bv = bias[h];
    float* obase = out + (size_t)bh * S_LEN * S_LEN;

    // ---- Band GEMM: 16 v-tiles of 16 columns (v = 0..255), K = 64 ----
    // Two independent accumulator chains (even/odd k) to break the WMMA RAW chain.
    v8f acc0 = {};
#pragma unroll 1
    for (int vt = 0; vt < 16; ++vt) {
        const int vbase = vt * 16;
        const int v  = vbase + row;              // this lane's B column
        const int sw = (v & 15) << 2;
        const float* bp = ldsE + v * 64;

        v8f accE = {};
        v8f accO = {};
#pragma unroll
        for (int kk = 0; kk < 8; ++kk) {
            const int ke = 2 * kk;
            const int ko = 2 * kk + 1;
            v2f be = *(const v2f*)(bp + ((4 * ke + koff) ^ sw));
            v2f bo = *(const v2f*)(bp + ((4 * ko + koff) ^ sw));
            accE = __builtin_amdgcn_wmma_f32_16x16x4_f32(
                false, a[ke], false, be, (short)0, accE, false, false);
            accO = __builtin_amdgcn_wmma_f32_16x16x4_f32(
                false, a[ko], false, bo, (short)0, accO, false, false);
        }
        v8f acc = accE + accO;
        if (vt == 0) acc0 = acc;                 // keep Y[s,0..15] for left-fill broadcast

        // C layout: vgpr r -> M = r + half*8, N = row. Write t = s + v - 128.
#pragma unroll
        for (int r = 0; r < 8; ++r) {
            int m = r + half * 8;
            int s = s0 + m;
            int t = s + vbase + row - MAXREL;
            if (t >= 0 && t < S_LEN)
                obase[(size_t)s * S_LEN + t] = acc[r] + bv;
        }
    }

    // ---- v = 256 single band position: t = s + 128 ----
    if (lane < 16) {
        int s = s0 + lane;
        if (s + MAXREL < S_LEN)
            obase[(size_t)s * S_LEN + (s + MAXREL)] = y256v + bv;
    }

    // ---- Broadcast fills: t < s-128 gets Y[s,0]+bias ; t > s+128 gets Y[s,256]+bias ----
#pragma unroll
    for (int m = 0; m < 16; ++m) {
        const int s = s0 + m;
        float* rowp = obase + (size_t)s * S_LEN;

        // Y[s,0] lives in acc0[m&7] of lane (m>>3)*16 ; y256[s] in lane m
        const float y0b = __shfl(acc0[m & 7], (m >> 3) * 16, 32) + bv;
        const float y2b = __shfl(y256v, m, 32) + bv;

        // Left region: t in [0, s-129], count = s-128
        int Lc = s - MAXREL;
        if (Lc > 0) {
            const float4 f0 = make_float4(y0b, y0b, y0b, y0b);
            int nv = Lc >> 2;
            for (int i = lane; i < nv; i += 32) ((float4*)rowp)[i] = f0;
            for (int t = (nv << 2) + lane; t < Lc; t += 32) rowp[t] = y0b;
        }

        // Right region: t in [s+129, 1023]
        int start = s + MAXREL + 1;
        if (start < S_LEN) {
            int a4 = (start + 3) & ~3;
            if (a4 > S_LEN) a4 = S_LEN;
            for (int t = start + lane; t < a4; t += 32) rowp[t] = y2b; // unaligned head
            const float4 f2 = make_float4(y2b, y2b, y2b, y2b);
            for (int i = (a4 >> 2) + lane; i < (S_LEN >> 2); i += 32)
                ((float4*)rowp)[i] = f2;
        }
    }
}

extern "C" void kernel_launch(void* const* d_in, const int* in_sizes, int n_in,
                              void* d_out, int out_size, void* d_ws, size_t ws_size,
                              hipStream_t stream) {
    const float* X    = (const float*)d_in[0]; // tensor [4,16,1024,64]
    const float* E    = (const float*)d_in[1]; // rel_keys_emb [257,64]
    // d_in[2] = rel_values_emb (unused by forward is_key path)
    const float* bias = (const float*)d_in[3]; // [1,16,1,1] -> 16 floats
    float* out = (float*)d_out;                // [4,16,1024,1024]

    dim3 grid(512), block(256);
    relpos_scores_kernel<<<grid, block, 0, stream>>>(X, E, bias, out);
}